// AdaptiveKSelector_76982993814145
// MI455X (gfx1250) — compile-verified
//
#include <hip/hip_runtime.h>
#include <hip/hip_bf16.h>

// AdaptiveKSelector for MI455X (gfx1250), wave32.
// One wave per (b,q) row: async-stage row to LDS, 4x8-bit radix-select the
// k-th largest (k = min(64, q+1)), then stream the 0/1 mask out with exact
// stable-tie handling (lowest index first among equals).

#define WAVES 4
#define SMAX 2048

// match the builtin's parameter type exactly: gcc-style vector of 4 ints
typedef int v4i_t __attribute__((vector_size(16)));

__device__ __forceinline__ unsigned toOrd(float f) {
  // monotonic float -> uint mapping (negatives flipped, positives +sign bit)
  unsigned u = __float_as_uint(f);
  unsigned m = (unsigned)(((int)u) >> 31) | 0x80000000u;
  return u ^ m;
}

__device__ __forceinline__ void dsbar() {
  // split-counter wait: all outstanding LDS ops complete (gfx1250)
  asm volatile("s_wait_dscnt 0" ::: "memory");
}

__device__ __forceinline__ unsigned ballot32(bool p) {
#if __has_builtin(__builtin_amdgcn_ballot_w32)
  return __builtin_amdgcn_ballot_w32(p);
#else
  return (unsigned)__ballot(p ? 1 : 0);
#endif
}

__global__ __launch_bounds__(WAVES * 32) void adaptive_k_topk_kernel(
    const float* __restrict__ scores, float* __restrict__ out,
    int S, int rows, int kf) {
  __shared__ __align__(16) float s_row[WAVES][SMAX];
  __shared__ unsigned s_hist[WAVES][256];
  __shared__ unsigned s_sums[WAVES][32];
  __shared__ unsigned s_bcast[WAVES][2];

  const int lane = threadIdx.x & 31;
  const int wv   = threadIdx.x >> 5;
  const int row  = blockIdx.x * WAVES + wv;
  if (row >= rows) return;  // no block-wide barriers used anywhere

  const int q = row % S;
  const int L = q + 1;                       // causal row length
  const int kEff = (kf < L) ? kf : L;        // min(k, q+1)
  const size_t gbase = (size_t)row * (size_t)S;

  float*    rowW  = s_row[wv];
  unsigned* histw = s_hist[wv];
  unsigned* sums  = s_sums[wv];
  unsigned* bcast = s_bcast[wv];

  if (lane == 0) {
    // k_values output (second tuple element), constant kf per row
    out[(size_t)rows * (size_t)S + (size_t)row] = (float)kf;
  }

  if (kEff >= L) {
    // k >= row length: every causal position selected (mask == causal row)
    for (int j = lane; j < S; j += 32)
      out[gbase + (size_t)j] = (j < L) ? 1.0f : 0.0f;
    return;
  }

  // ---------------- stage row into LDS ----------------
  const bool okAsync = ((S & 127) == 0) && (L <= SMAX);
#if defined(__AMDGCN__) && __has_builtin(__builtin_amdgcn_global_load_async_to_lds_b128)
  if (okAsync) {
    const char* gsrc = (const char*)(scores + gbase);
    char* ldst = (char*)rowW;
    const int chunks = (L * 4 + 511) >> 9;   // 512B per wave iteration
    for (int i = 0; i < chunks; ++i) {
      const int off = (i * 32 + lane) * 16;  // 16B per lane, coalesced
      __builtin_amdgcn_global_load_async_to_lds_b128(
          (__attribute__((address_space(1))) v4i_t*)(gsrc + off),
          (__attribute__((address_space(3))) v4i_t*)(ldst + off),
          0, 0);
    }
    asm volatile("s_wait_asynccnt 0" ::: "memory");
  } else
#endif
  {
    (void)okAsync;
    for (int j = lane; j < L; j += 32) rowW[j] = scores[gbase + (size_t)j];
  }
  dsbar();

  // ---------------- 4-pass radix select (MSB-first, 8-bit digits) ----------
  unsigned long long prefHigh = 0ull;   // decided high bits of threshold key
  unsigned remK = (unsigned)kEff;       // rank still needed inside candidates
  for (int p = 3; p >= 0; --p) {
    const int shift = 8 * p;
    for (int i = 0; i < 8; ++i) histw[lane * 8 + i] = 0u;
    dsbar();
    for (int j = lane; j < L; j += 32) {
      const unsigned key = toOrd(rowW[j]);
      if (((unsigned long long)key >> (shift + 8)) == prefHigh) {
        atomicAdd(&histw[(key >> shift) & 0xFFu], 1u);
      }
    }
    dsbar();
    // lane owns bins [8*lane, 8*lane+7]; count above-my-range via LDS partials
    unsigned P = 0;
    for (int i = 0; i < 8; ++i) P += histw[lane * 8 + i];
    sums[lane] = P;
    dsbar();
    unsigned T = 0;  // candidates with digit strictly above my bin range
    for (int l2 = lane + 1; l2 < 32; ++l2) T += sums[l2];
    const bool hit = (T < remK) && (remK <= T + P);   // unique lane
    const unsigned ball = ballot32(hit);
    const int sl = __ffs(ball) - 1;
    if (lane == sl) {
      unsigned cum = T;
      for (int d = lane * 8 + 7;; --d) {
        const unsigned c = histw[d];
        if (cum + c >= remK) { bcast[0] = (unsigned)d; bcast[1] = remK - cum; break; }
        cum += c;
      }
    }
    dsbar();
    prefHigh = (prefHigh << 8) | (unsigned long long)bcast[0];
    remK = bcast[1];
  }
  const unsigned thresh = (unsigned)prefHigh;  // exact key of k-th largest
  const unsigned R = remK;                     // #equal-to-threshold to keep (>=1)

  // ---------------- emit mask, stable tie-break by lowest index ------------
  unsigned eqSeen = 0;
  const unsigned ltMask = (1u << lane) - 1u;
  for (int b0 = 0; b0 < S; b0 += 32) {
    const int j = b0 + lane;
    const bool valid = (j < L);
    const unsigned key = valid ? toOrd(rowW[j]) : 0u;
    const bool eq = valid && (key == thresh);
    const unsigned bal = ballot32(eq);
    const bool sel = (valid && key > thresh) ||
                     (eq && (eqSeen + (unsigned)__popc(bal & ltMask)) < R);
    if (j < S) out[gbase + (size_t)j] = sel ? 1.0f : 0.0f;
    eqSeen += (unsigned)__popc(bal);
  }
}

extern "C" void kernel_launch(void* const* d_in, const int* in_sizes, int n_in,
                              void* d_out, int out_size, void* d_ws, size_t ws_size,
                              hipStream_t stream) {
  (void)n_in; (void)d_ws; (void)ws_size;
  // inputs: d_in[0]=x (unused), d_in[1]=index_scores [B,S,S] f32
  const float* scores = (const float*)d_in[1];
  const long long T  = (long long)in_sizes[1];       // B*S*S
  const long long BS = (long long)out_size - T;      // B*S
  const int S    = (int)(T / BS);
  const int rows = (int)BS;

  // k_fixed = clip(min(64,S), 16, min(512,S)) (lower then upper, like np.clip)
  int kf = (S < 64) ? S : 64;
  if (kf < 16) kf = 16;
  const int hi = (S < 512) ? S : 512;
  if (kf > hi) kf = hi;

  dim3 block(WAVES * 32);
  dim3 grid((rows + WAVES - 1) / WAVES);
  adaptive_k_topk_kernel<<<grid, block, 0, stream>>>(scores, (float*)d_out,
                                                     S, rows, kf);
}